// Ptr_Layer_22797686408009
// MI455X (gfx1250) — compile-verified
//
#include <hip/hip_runtime.h>
#include <hip/hip_bf16.h>

// Problem constants (from reference): B=4, S=512, F=256, A=128
constexpr int BB = 4;
constexpr int SS = 512;
constexpr int FF = 256;
constexpr int AA = 128;

typedef __attribute__((ext_vector_type(2))) float v2f;
typedef __attribute__((ext_vector_type(8))) float v8f;

// Hardware tanh (V_TANH_F32) when available on gfx1250; libm fallback otherwise.
__device__ inline float fast_tanh(float x) {
#if defined(__has_builtin)
#if __has_builtin(__builtin_amdgcn_tanhf)
    return __builtin_amdgcn_tanhf(x);
#else
    return tanhf(x);
#endif
#else
    return tanhf(x);
#endif
}

// ---------------------------------------------------------------------------
// Generic 16x16 fp32 tile GEMM using V_WMMA_F32_16X16X4_F32 (wave32).
// C[16x16] += A[16xK] * B[Kx16], K stepped by 4.
// A element (m,k) at Abase[m*A_rs + k*A_cs]
// B element (k,n) at Bbase[k*B_rs + n*B_cs]
// ISA layout (32-bit A 16x4): lanes 0-15 hold K={0,1}, lanes 16-31 hold K={2,3}
// ---------------------------------------------------------------------------
__device__ inline v8f wmma_tile_f32(const float* __restrict__ Abase, int A_rs, int A_cs,
                                    const float* __restrict__ Bbase, int B_rs, int B_cs,
                                    int K)
{
    const int lane = threadIdx.x & 31;
    const int half = lane >> 4;   // 0 -> K slots {0,1}; 1 -> K slots {2,3}
    const int lrow = lane & 15;   // M index for A fragment, N index for B fragment
    v8f c = {};
#pragma unroll 8
    for (int k0 = 0; k0 < K; k0 += 4) {
        const int ka = k0 + 2 * half;
        v2f a, b;
        a.x = Abase[lrow * A_rs + (ka + 0) * A_cs];
        a.y = Abase[lrow * A_rs + (ka + 1) * A_cs];
        b.x = Bbase[(ka + 0) * B_rs + lrow * B_cs];
        b.y = Bbase[(ka + 1) * B_rs + lrow * B_cs];
        c = __builtin_amdgcn_wmma_f32_16x16x4_f32(false, a, false, b,
                                                  (short)0, c, false, false);
    }
    return c;
}

// C/D 16x16 f32 layout: VGPR r, lanes 0-15 -> M=r, N=lane; lanes 16-31 -> M=r+8.
__device__ inline void store_tile_f32(float* __restrict__ Cbase, int C_rs, v8f c)
{
    const int lane = threadIdx.x & 31;
    const int half = lane >> 4;
    const int lrow = lane & 15;
#pragma unroll
    for (int r = 0; r < 8; ++r) {
        Cbase[(r + 8 * half) * C_rs + lrow] = c[r];
    }
}

// ---------------------------------------------------------------------------
// Kernel 1: projections e = e_T @ W1^T, d = d_T @ W2^T   ([B,S,F] x [A,F] -> [B,S,A])
// grid.x = 128 blocks of 8 waves -> 1024 tiles per matrix; grid.y selects e vs d.
// ---------------------------------------------------------------------------
__global__ __launch_bounds__(256) void proj_kernel(const float* __restrict__ eT,
                                                   const float* __restrict__ dT,
                                                   const float* __restrict__ W1,
                                                   const float* __restrict__ W2,
                                                   float* __restrict__ e_out,
                                                   float* __restrict__ d_out)
{
    const float* X = (blockIdx.y == 0) ? eT : dT;
    const float* W = (blockIdx.y == 0) ? W1 : W2;
    float*       O = (blockIdx.y == 0) ? e_out : d_out;

    const int wave  = blockIdx.x * (blockDim.x >> 5) + (threadIdx.x >> 5); // 0..1023
    const int atile = wave & 7;          // A/16 = 8
    const int stile = (wave >> 3) & 31;  // S/16 = 32
    const int b     = wave >> 8;         // 0..3

    // A(m,k) = X[b, stile*16+m, k]          -> rs=FF, cs=1
    // B(k,n) = W[atile*16+n, k]             -> rs=1,  cs=FF
    const float* Abase = X + ((size_t)b * SS + (size_t)stile * 16) * FF;
    const float* Bbase = W + (size_t)atile * 16 * FF;
    v8f c = wmma_tile_f32(Abase, FF, 1, Bbase, 1, FF, FF);

    float* Cbase = O + ((size_t)b * SS + (size_t)stile * 16) * AA + atile * 16;
    store_tile_f32(Cbase, AA, c);
}

// ---------------------------------------------------------------------------
// Kernel 2: ut[b,j,i] = softmax_j( sum_a v[a]*tanh(e[b,i,a]+d[b,j,a]) )
// One block (256 threads) per (b,i) row; each thread covers j and j+256.
// e-row and v live in LDS; d[b] rows stream from L2 as float4.
// Output written TRANSPOSED (ut[b,j,i]) so the final GEMM reads K-contiguous.
// ---------------------------------------------------------------------------
__global__ __launch_bounds__(256) void score_softmax_kernel(const float* __restrict__ e,
                                                            const float* __restrict__ d,
                                                            const float* __restrict__ v,
                                                            float* __restrict__ ut)
{
    __shared__ float se[AA];
    __shared__ float sv[AA];
    __shared__ float red[256];

    const int b = blockIdx.x / SS;
    const int i = blockIdx.x % SS;

    const float* erow = e + ((size_t)b * SS + i) * AA;
    for (int t = threadIdx.x; t < AA; t += blockDim.x) {
        se[t] = erow[t];
        sv[t] = v[t];
    }
    __syncthreads();

    const float* dbase = d + (size_t)b * SS * AA;
    const float4* e4 = (const float4*)se;
    const float4* v4 = (const float4*)sv;

    float uv[2];
#pragma unroll
    for (int jj = 0; jj < 2; ++jj) {
        const int j = threadIdx.x + jj * 256;
        const float4* d4 = (const float4*)(dbase + (size_t)j * AA);
        float acc = 0.0f;
#pragma unroll 8
        for (int a4 = 0; a4 < AA / 4; ++a4) {
            const float4 dv = d4[a4];
            const float4 ev = e4[a4];
            const float4 vv = v4[a4];
            acc = fmaf(vv.x, fast_tanh(ev.x + dv.x), acc);
            acc = fmaf(vv.y, fast_tanh(ev.y + dv.y), acc);
            acc = fmaf(vv.z, fast_tanh(ev.z + dv.z), acc);
            acc = fmaf(vv.w, fast_tanh(ev.w + dv.w), acc);
        }
        uv[jj] = acc;
    }

    // --- block-wide max over 512 values ---
    red[threadIdx.x] = fmaxf(uv[0], uv[1]);
    __syncthreads();
    for (int off = 128; off > 0; off >>= 1) {
        if (threadIdx.x < off)
            red[threadIdx.x] = fmaxf(red[threadIdx.x], red[threadIdx.x + off]);
        __syncthreads();
    }
    const float m = red[0];
    __syncthreads();

    const float e0 = __expf(uv[0] - m);
    const float e1 = __expf(uv[1] - m);

    // --- block-wide sum ---
    red[threadIdx.x] = e0 + e1;
    __syncthreads();
    for (int off = 128; off > 0; off >>= 1) {
        if (threadIdx.x < off)
            red[threadIdx.x] += red[threadIdx.x + off];
        __syncthreads();
    }
    const float inv = 1.0f / red[0];

    // transposed store: ut[b, j, i]
    float* ub = ut + (size_t)b * SS * SS;
    ub[(size_t)(threadIdx.x)       * SS + i] = e0 * inv;
    ub[(size_t)(threadIdx.x + 256) * SS + i] = e1 * inv;
}

// ---------------------------------------------------------------------------
// Kernel 3: out[b,j,f] = sum_i ut[b,j,i] * x_T[b,i,f]   (K-contiguous A reads)
// 2048 tiles of 16x16, one per wave, K = S = 512.
// ---------------------------------------------------------------------------
__global__ __launch_bounds__(256) void out_kernel(const float* __restrict__ ut,
                                                  const float* __restrict__ x,
                                                  float* __restrict__ out)
{
    const int wave  = blockIdx.x * (blockDim.x >> 5) + (threadIdx.x >> 5); // 0..2047
    const int ftile = wave & 15;         // F/16 = 16
    const int jtile = (wave >> 4) & 31;  // S/16 = 32
    const int b     = wave >> 9;         // 0..3

    // A(m,k) = ut[b, jtile*16+m, k] -> rs=SS, cs=1  (contiguous K)
    // B(k,n) = x[b, k, ftile*16+n]  -> rs=FF, cs=1
    const float* Abase = ut + ((size_t)b * SS + (size_t)jtile * 16) * SS;
    const float* Bbase = x + (size_t)b * SS * FF + (size_t)ftile * 16;
    v8f c = wmma_tile_f32(Abase, SS, 1, Bbase, FF, 1, SS);

    float* Cbase = out + ((size_t)b * SS + (size_t)jtile * 16) * FF + ftile * 16;
    store_tile_f32(Cbase, FF, c);
}

// ---------------------------------------------------------------------------
extern "C" void kernel_launch(void* const* d_in, const int* in_sizes, int n_in,
                              void* d_out, int out_size, void* d_ws, size_t ws_size,
                              hipStream_t stream)
{
    const float* x_T = (const float*)d_in[0];
    const float* e_T = (const float*)d_in[1];
    const float* d_T = (const float*)d_in[2];
    const float* W1  = (const float*)d_in[3];
    const float* W2  = (const float*)d_in[4];
    const float* v   = (const float*)d_in[5];
    float* out = (float*)d_out;

    // Workspace: e [B,S,A] | d [B,S,A] | ut [B,S,S]  = 1MB + 1MB + 4MB
    float* e_ws   = (float*)d_ws;
    float* d_ws_p = e_ws + (size_t)BB * SS * AA;
    float* u_ws   = d_ws_p + (size_t)BB * SS * AA;

    // 1) projections (both e and d via grid.y)
    proj_kernel<<<dim3(128, 2), 256, 0, stream>>>(e_T, d_T, W1, W2, e_ws, d_ws_p);

    // 2) tanh scores + softmax over j (writes u transposed)
    score_softmax_kernel<<<BB * SS, 256, 0, stream>>>(e_ws, d_ws_p, v, u_ws);

    // 3) out = u^T @ x per batch
    out_kernel<<<256, 256, 0, stream>>>(u_ws, x_T, out);
}